// RankDepthLoss_49065706390299
// MI455X (gfx1250) — compile-verified
//
#include <hip/hip_runtime.h>
#include <hip/hip_bf16.h>

typedef __attribute__((ext_vector_type(2))) float v2f;
typedef __attribute__((ext_vector_type(8))) float v8f;

#define S_LEN   1024
#define B_SIZE  64
#define NSPLIT  4      // blocks per batch row
#define NWAVES  8      // waves per block (256 threads, wave32)
#define NTILES  64     // S_LEN / 16

// Pairwise rank-hinge over the tile triangle of one batch row.
// pred-diff and label-diff 16x16 tiles come from V_WMMA_F32_16X16X4_F32
// (outer difference p_i*1 + 1*(-p_j), bit-exact vs v_sub). Labels are
// integer-valued, so sign(ldiff) == med3(ldiff,-1,1) (single v_med3, no
// canonicalize), and the hinge is med3/fma/max/pk_fma only, co-executing
// with the WMMA pipe. 8 independent accumulators avoid serial add chains.
__global__ __launch_bounds__(256)
void RankDepthLoss_pair_kernel(const float* __restrict__ pred,
                               const float* __restrict__ label,
                               float* __restrict__ partials) {
    // Packed row: {p_masked, l_masked, mask, 0} -> one ds_load per tile.
    __shared__ float4 srow[S_LEN];
    __shared__ float  red[NWAVES];

    const int b     = blockIdx.x / NSPLIT;
    const int split = blockIdx.x % NSPLIT;
    const int tid   = threadIdx.x;

    for (int k = tid; k < S_LEN; k += 256) {
        float lb = label[b * S_LEN + k];
        float m  = (lb != -1.0f) ? 1.0f : 0.0f;
        float4 e;
        e.x = pred[b * S_LEN + k] * m;   // p
        e.y = lb * m;                    // l (integer-valued)
        e.z = m;                         // mask
        e.w = 0.0f;
        srow[k] = e;
    }
    __syncthreads();

    const int wave = tid >> 5;
    const int lane = tid & 31;
    const int half = (lane >= 16) ? 1 : 0;   // high lanes hold K=2,3 of A/B
    const int l15  = lane & 15;

    // Per-lane diagonal mask: D elem r maps to (M = r + 8*half, N = l15);
    // diagonal tiles keep only M < N.
    float dmask[8];
#pragma unroll
    for (int r = 0; r < 8; ++r)
        dmask[r] = ((r + half * 8) < l15) ? 1.0f : 0.0f;

    const int v = split * NWAVES + wave;     // 0..31: tile-rows v and 63-v
    float acc[8];
#pragma unroll
    for (int r = 0; r < 8; ++r) acc[r] = 0.0f;

    for (int pass = 0; pass < 2; ++pass) {
        const int ti = pass ? (NTILES - 1 - v) : v;
        const int I  = ti * 16;

        // A tiles (16x4 f32), built once per pass. K=2,3 (high lanes) are
        // exactly zero, so B's high-lane contents are irrelevant (finite*0).
        const float4 di = srow[I + l15];
        v2f pa, la;
        pa.x = half ? 0.0f : di.x;   // K0 = p_i   | K2 = 0
        pa.y = half ? 0.0f : 1.0f;   // K1 = 1     | K3 = 0
        la.x = half ? 0.0f : di.y;   // K0 = l_i
        la.y = pa.y;

        // One-deep software pipeline on the packed tile load.
        float4 dj = srow[ti * 16 + l15];
        for (int tj = ti; tj < NTILES; ++tj) {
            const float4 cur = dj;
            const int tjn = (tj + 1 < NTILES) ? (tj + 1) : tj;
            dj = srow[tjn * 16 + l15];   // overlaps with WMMA + hinge VALU

            // B tiles (4x16 f32): unconditional, no lane selects needed.
            v2f pb, lb;
            pb.x = 1.0f;        // row K0 = 1
            pb.y = -cur.x;      // row K1 = -p_j
            lb.x = 1.0f;
            lb.y = -cur.y;

            v8f zc = {};
            // D[m][n] = p[I+m] - p[J+n]   (exact)
            v8f pd = __builtin_amdgcn_wmma_f32_16x16x4_f32(
                false, pa, false, pb, (short)0, zc, false, false);
            // D[m][n] = l[I+m] - l[J+n]   (exact small integers)
            v8f ld = __builtin_amdgcn_wmma_f32_16x16x4_f32(
                false, la, false, lb, (short)0, zc, false, false);

            const float mj = cur.z;      // mask_j (N = l15 for every reg)

            if (ti == tj) {
                // diagonal tile: fold i<j predicate into the mask
#pragma unroll
                for (int r = 0; r < 8; ++r) {
                    float s = __builtin_amdgcn_fmed3f(ld[r], -1.0f, 1.0f);
                    float h = fmaxf(fmaf(-s, pd[r], 1.0f), 0.0f);
                    acc[r] = fmaf(h, mj * dmask[r], acc[r]);
                }
            } else {
#pragma unroll
                for (int r = 0; r < 8; ++r) {
                    float s = __builtin_amdgcn_fmed3f(ld[r], -1.0f, 1.0f);
                    float h = fmaxf(fmaf(-s, pd[r], 1.0f), 0.0f);
                    acc[r] = fmaf(h, mj, acc[r]);               // independent
                }
            }
        }
    }

    // Collapse the 8 accumulators, then deterministic wave32 tree reduction.
    float a = ((acc[0] + acc[1]) + (acc[2] + acc[3]))
            + ((acc[4] + acc[5]) + (acc[6] + acc[7]));
    for (int off = 16; off > 0; off >>= 1)
        a += __shfl_down(a, off, 32);
    if (lane == 0) red[wave] = a;
    __syncthreads();

    if (tid == 0) {
        float t = 0.0f;
        for (int w = 0; w < NWAVES; ++w) t += red[w];
        partials[b * NSPLIT + split] = t;   // plain store: no atomics
    }
}

// Combine per-block partials: rank_loss[b]/len_b^2, sum over b, / total_sents.
__global__ void RankDepthLoss_finalize_kernel(const float* __restrict__ partials,
                                              const int* __restrict__ lengths,
                                              float* __restrict__ out) {
    const int lane = threadIdx.x;   // 32 threads
    float lsum = 0.0f, cnt = 0.0f;
    for (int b = lane; b < B_SIZE; b += 32) {
        float rsum = 0.0f;
        for (int s = 0; s < NSPLIT; ++s) rsum += partials[b * NSPLIT + s];
        float len = (float)lengths[b];
        lsum += rsum / (len * len);
        cnt  += (lengths[b] != 0) ? 1.0f : 0.0f;
    }
    for (int off = 16; off > 0; off >>= 1) {
        lsum += __shfl_down(lsum, off, 32);
        cnt  += __shfl_down(cnt,  off, 32);
    }
    if (lane == 0) {
        out[0] = lsum / cnt;
        out[1] = cnt;
    }
}

extern "C" void kernel_launch(void* const* d_in, const int* in_sizes, int n_in,
                              void* d_out, int out_size, void* d_ws, size_t ws_size,
                              hipStream_t stream) {
    (void)in_sizes; (void)n_in; (void)out_size; (void)ws_size;
    const float* pred    = (const float*)d_in[0];   // [64,1024] f32
    const float* label   = (const float*)d_in[1];   // [64,1024] f32
    const int*   lengths = (const int*)d_in[2];     // [64] i32
    float* out      = (float*)d_out;                // [2] f32
    float* partials = (float*)d_ws;                 // 64*NSPLIT floats

    RankDepthLoss_pair_kernel<<<B_SIZE * NSPLIT, 256, 0, stream>>>(pred, label, partials);
    RankDepthLoss_finalize_kernel<<<1, 32, 0, stream>>>(partials, lengths, out);
}